// Distorted_SSIM_33655363731664
// MI455X (gfx1250) — compile-verified
//
#include <hip/hip_runtime.h>
#include <hip/hip_bf16.h>

// ---------------------------------------------------------------------------
// Multi-window SSIM, fully fused, CDNA5 (gfx1250) wave32.
// - Vertical Gaussian pass on V_WMMA_F32_16X16X4_F32 (banded-matrix GEMM)
// - Horizontal pass in packed VALU from LDS
// - Halo ingest via GLOBAL_LOAD_ASYNC_TO_LDS (ASYNCcnt) when toolchain has it
// - SSIM map + mean fused, double atomic reduction
// ---------------------------------------------------------------------------

typedef __attribute__((ext_vector_type(2))) float v2f;
typedef __attribute__((ext_vector_type(8))) float v8f;

#define HH 512
#define WW 512
#define TS 16
#define HALO 5
#define LROWS 26   // TS + 2*HALO
#define LSTRIDE 28 // padded LDS row stride (floats)

#if __has_builtin(__builtin_amdgcn_global_load_async_to_lds_b32)
#define SSIM_ASYNC_LDS 1
typedef __attribute__((address_space(1))) int *gas_i32_ptr;
typedef __attribute__((address_space(3))) int *las_i32_ptr;
#endif

static __device__ __forceinline__ v8f wmma4(v2f a, v2f b, v8f c) {
  // D(16x16,f32) = A(16x4,f32) x B(4x16,f32) + C
  return __builtin_amdgcn_wmma_f32_16x16x4_f32(
      /*neg_a=*/false, a, /*neg_b=*/false, b,
      /*c_mod=*/(short)0, c, /*reuse_a=*/false, /*reuse_b=*/false);
}

// 1-D Gaussian taps, sigma = 1.5, normalized (precomputed from the reference).
template <int N> static __device__ __forceinline__ float gauss1d(int i) {
  if constexpr (N == 5) {
    constexpr float g[5] = {0.1200789f, 0.2338813f, 0.2920796f,
                            0.2338813f, 0.1200789f};
    return g[i];
  } else {
    constexpr float g[11] = {0.00102845f, 0.00759873f, 0.03600077f,
                             0.10936072f, 0.21300543f, 0.26601170f,
                             0.21300543f, 0.10936072f, 0.03600077f,
                             0.00759873f, 0.00102845f};
    return g[i];
  }
}

// Build banded vertical-weight matrix W[m][k] = gv[k - m] (0 <= k-m < KH).
template <int KH>
static __device__ __forceinline__ void build_band(float (&wb)[TS][LSTRIDE],
                                                  int lane) {
  float *w = &wb[0][0];
  for (int i = lane; i < TS * LSTRIDE; i += 32) w[i] = 0.0f;
  __syncthreads();
  if (lane < TS) {
#pragma unroll
    for (int t = 0; t < KH; ++t) wb[lane][lane + t] = gauss1d<KH>(t);
  }
  __syncthreads();
}

// Per-lane partial sum of the SSIM map over one 16x16 tile, one window shape.
// wb must already hold the KH band.
template <int KH, int KW>
static __device__ float ssim_tile(const float (&s1)[LROWS][LSTRIDE],
                                  const float (&s2)[LROWS][LSTRIDE],
                                  const float (&wb)[TS][LSTRIDE], int lane) {
  constexpr int PH = KH / 2;
  constexpr int PW = KW / 2;
  constexpr int R = TS + KH - 1;   // rows in vertical band
  constexpr int NCH = (R + 3) / 4; // K=4 chunks per WMMA accumulation
  constexpr float C1 = 1.0e-4f;    // 0.01^2
  constexpr float C2 = 9.0e-4f;    // 0.03^2

  const int m = lane & 15;    // A-matrix row / B-matrix column (N)
  const int half = lane >> 4; // which K pair this lane holds

  v8f a0 = {}, a1 = {}, a2 = {}, a3 = {}, a4 = {};

#pragma unroll
  for (int ch = 0; ch < NCH; ++ch) {
    const int k0 = 4 * ch + 2 * half;
    v2f A, B0, B1, B2, B3, B4;
#pragma unroll
    for (int v = 0; v < 2; ++v) {
      const int k = k0 + v;
      A[v] = wb[m][k];                      // zero beyond band -> safe
      const int kk = (k < R) ? k : (R - 1); // clamp data row (weight is 0)
      const float *ar = &s1[(HALO - PH) + kk][(HALO - PW) + m];
      const float *br = &s2[(HALO - PH) + kk][(HALO - PW) + m];
      float h0 = 0.f, h1 = 0.f, h2 = 0.f, h3 = 0.f, h4 = 0.f;
#pragma unroll
      for (int j = 0; j < KW; ++j) { // horizontal Gaussian (packed VALU)
        const float g = gauss1d<KW>(j);
        const float x = ar[j];
        const float y = br[j];
        h0 += g * x;
        h1 += g * y;
        h2 += g * x * x;
        h3 += g * y * y;
        h4 += g * x * y;
      }
      B0[v] = h0; B1[v] = h1; B2[v] = h2; B3[v] = h3; B4[v] = h4;
    }
    a0 = wmma4(A, B0, a0);
    a1 = wmma4(A, B1, a1);
    a2 = wmma4(A, B2, a2);
    a3 = wmma4(A, B3, a3);
    a4 = wmma4(A, B4, a4);
  }

  // Pointwise SSIM on the 8 accumulator elements this lane owns (layout-
  // agnostic: all five accumulators share the C/D layout).
  float s = 0.f;
#pragma unroll
  for (int e = 0; e < 8; ++e) {
    const float mu1 = a0[e];
    const float mu2 = a1[e];
    const float m11 = mu1 * mu1;
    const float m22 = mu2 * mu2;
    const float m12 = mu1 * mu2;
    const float sg1 = a2[e] - m11;
    const float sg2 = a3[e] - m22;
    const float sg12 = a4[e] - m12;
    s += ((2.f * m12 + C1) * (2.f * sg12 + C2)) /
         ((m11 + m22 + C1) * (sg1 + sg2 + C2));
  }
  return s;
}

// Zero-padded 26x26 halo load of both images into LDS.
static __device__ __forceinline__ void
load_halo(const float *__restrict__ p1, const float *__restrict__ p2,
          float (&s1)[LROWS][LSTRIDE], float (&s2)[LROWS][LSTRIDE], int r0,
          int c0, int lane) {
  const int gc = c0 - HALO + lane;
  const bool cok = (lane < LROWS) && ((unsigned)gc < (unsigned)WW);
#if defined(SSIM_ASYNC_LDS)
  // Border tiles: pre-zero LDS so masked-off (out-of-image) lanes read 0.
  const bool border =
      (r0 < HALO) || (c0 < HALO) || (r0 + TS + HALO > HH) || (c0 + TS + HALO > WW);
  if (border) {
    float *w1 = &s1[0][0];
    float *w2 = &s2[0][0];
    for (int i = lane; i < LROWS * LSTRIDE; i += 32) {
      w1[i] = 0.0f;
      w2[i] = 0.0f;
    }
    __syncthreads(); // DS stores retire before async writes arrive
  }
  for (int r = 0; r < LROWS; ++r) {
    const int gr = r0 - HALO + r;
    if (cok && ((unsigned)gr < (unsigned)HH)) {
      const size_t off = (size_t)gr * WW + gc;
      __builtin_amdgcn_global_load_async_to_lds_b32(
          (gas_i32_ptr)(p1 + off), (las_i32_ptr)&s1[r][lane], 0, 0);
      __builtin_amdgcn_global_load_async_to_lds_b32(
          (gas_i32_ptr)(p2 + off), (las_i32_ptr)&s2[r][lane], 0, 0);
    }
  }
#if __has_builtin(__builtin_amdgcn_s_wait_asynccnt)
  __builtin_amdgcn_s_wait_asynccnt(0);
#else
  asm volatile("s_wait_asynccnt 0x0" ::: "memory");
#endif
#else
  for (int r = 0; r < LROWS; ++r) {
    const int gr = r0 - HALO + r;
    const bool ok = cok && ((unsigned)gr < (unsigned)HH);
    const size_t off = (size_t)gr * WW + gc;
    if (lane < LROWS) {
      s1[r][lane] = ok ? p1[off] : 0.0f;
      s2[r][lane] = ok ? p2[off] : 0.0f;
    }
  }
#endif
}

__global__ __launch_bounds__(32) void ssim_main(const float *__restrict__ img1,
                                                const float *__restrict__ img2,
                                                double *__restrict__ acc) {
  __shared__ float s1[LROWS][LSTRIDE];
  __shared__ float s2[LROWS][LSTRIDE];
  __shared__ float wb[TS][LSTRIDE];

  const int lane = threadIdx.x;
  const int tile = blockIdx.x; // 96 planes * 32 * 32 tiles
  const int plane = tile >> 10;
  const int ty = (tile >> 5) & 31;
  const int tx = tile & 31;
  const int r0 = ty * TS;
  const int c0 = tx * TS;

  const float *p1 = img1 + (size_t)plane * HH * WW;
  const float *p2 = img2 + (size_t)plane * HH * WW;

  load_halo(p1, p2, s1, s2, r0, c0, lane);
  __syncthreads();

  float part = 0.f;
  build_band<5>(wb, lane);
  part += ssim_tile<5, 11>(s1, s2, wb, lane);
  build_band<11>(wb, lane); // shared by the two KH=11 windows
  part += ssim_tile<11, 5>(s1, s2, wb, lane);
  part += ssim_tile<11, 11>(s1, s2, wb, lane);

  // wave32 reduction
#pragma unroll
  for (int o = 16; o > 0; o >>= 1) part += __shfl_xor(part, o, 32);
  if (lane == 0) atomicAdd(acc, (double)part);
}

__global__ void ssim_init(double *acc) { *acc = 0.0; }

__global__ void ssim_fin(const double *acc, float *out) {
  // mean over 3 windows, each map has 32*3*512*512 elements
  constexpr double NMAP = 32.0 * 3.0 * 512.0 * 512.0;
  out[0] = (float)(*acc / (3.0 * NMAP));
}

extern "C" void kernel_launch(void *const *d_in, const int *in_sizes, int n_in,
                              void *d_out, int out_size, void *d_ws,
                              size_t ws_size, hipStream_t stream) {
  const float *img1 = (const float *)d_in[0];
  const float *img2 = (const float *)d_in[1];
  double *acc = (double *)d_ws;
  float *out = (float *)d_out;

  constexpr int PLANES = 32 * 3;
  constexpr int TILES = PLANES * (HH / TS) * (WW / TS); // 98304

  ssim_init<<<1, 1, 0, stream>>>(acc);
  ssim_main<<<TILES, 32, 0, stream>>>(img1, img2, acc);
  ssim_fin<<<1, 1, 0, stream>>>(acc, out);
}